// GIN_DDI_64622077935659
// MI455X (gfx1250) — compile-verified
//
#include <hip/hip_runtime.h>

typedef __attribute__((ext_vector_type(2))) float v2f;
typedef __attribute__((ext_vector_type(8))) float v8f;

#define N_NODES  50000
#define N_EDGES  800000
#define DIM      64
#define N_GRAPHS 1024
#define KGE_DIM  128
#define BN_EPS   1e-5f

__global__ void zero_kernel(float* __restrict__ p, int n) {
  int i = blockIdx.x * blockDim.x + threadIdx.x;
  if (i < n) p[i] = 0.f;
}

// agg[dst] += x[src] ; thread = (edge, 4-float chunk)
__global__ void edge_agg_kernel(const float* __restrict__ x,
                                const int* __restrict__ ei,
                                float* __restrict__ agg) {
  int gid = blockIdx.x * blockDim.x + threadIdx.x;
  if (gid >= N_EDGES * 16) return;
  int e = gid >> 4;
  int c = (gid & 15) * 4;
  int src = ei[e];
  int dst = ei[N_EDGES + e];
  const float4 v = *(const float4*)(x + (size_t)src * DIM + c);
  float* a = agg + (size_t)dst * DIM + c;
  atomicAdd(a + 0, v.x);
  atomicAdd(a + 1, v.y);
  atomicAdd(a + 2, v.z);
  atomicAdd(a + 3, v.w);
}

// Per-node MLP with FP32 WMMA (16x16x4, 16 K-steps over K=64).
// Block = 128 threads = 4 wave32, 16-node tile per wave, 64 nodes per block.
// Weights are async-copied global->LDS (ASYNCcnt) while the x+agg tile is
// staged through VGPRs (needs the add), overlapping the two.
__launch_bounds__(128)
__global__ void node_mlp_kernel(const float* __restrict__ x,
                                const float* __restrict__ agg,
                                const float* __restrict__ W1,   // 64x64 row-major (in,out)
                                const float* __restrict__ b1,
                                const float* __restrict__ W2,   // 64x64
                                const float* __restrict__ b2,
                                const float* __restrict__ gamma,
                                const float* __restrict__ beta,
                                const float* __restrict__ mean,
                                const float* __restrict__ var,
                                float* __restrict__ xout) {
  __shared__ __align__(16) float lds_tile[4 * 16 * DIM];   // 4 wave slices, 16 rows x 64
  __shared__ __align__(16) float lds_w[2 * DIM * DIM];     // W1 then W2

  const int tid = threadIdx.x;
  const int wave = tid >> 5;
  const int lane = tid & 31;
  const int block_node0 = blockIdx.x * 64;

  // ---- async weight copy: GLOBAL_LOAD_ASYNC_TO_LDS_B128 (LDS aperture keeps
  //      the workgroup-relative LDS offset in addr[31:0]) ----
  {
    const unsigned lw = (unsigned)(size_t)(void*)lds_w;
    for (int i = tid; i < DIM * DIM / 4; i += 128) {
      unsigned loff = lw + (unsigned)(i * 16);
      unsigned goff = (unsigned)(i * 16);
      asm volatile("global_load_async_to_lds_b128 %0, %1, %2"
                   :: "v"(loff), "v"(goff), "s"(W1) : "memory");
    }
    const unsigned lw2 = lw + DIM * DIM * 4;
    for (int i = tid; i < DIM * DIM / 4; i += 128) {
      unsigned loff = lw2 + (unsigned)(i * 16);
      unsigned goff = (unsigned)(i * 16);
      asm volatile("global_load_async_to_lds_b128 %0, %1, %2"
                   :: "v"(loff), "v"(goff), "s"(W2) : "memory");
    }
  }

  // ---- stage (x + agg) tile through VGPRs (coalesced float4), overlapping
  //      with the async weight copy ----
  for (int i = tid; i < 64 * DIM / 4; i += 128) {
    int node = block_node0 + (i >> 4);
    int cc = i & 15;
    float4 v = make_float4(0.f, 0.f, 0.f, 0.f);
    if (node < N_NODES) {
      float4 xv = ((const float4*)(x + (size_t)node * DIM))[cc];
      float4 av = ((const float4*)(agg + (size_t)node * DIM))[cc];
      v = make_float4(xv.x + av.x, xv.y + av.y, xv.z + av.z, xv.w + av.w);
    }
    ((float4*)lds_tile)[i] = v;
  }
  asm volatile("s_wait_asynccnt 0" ::: "memory");
  __syncthreads();

  float* slice = lds_tile + wave * 16 * DIM;
  const int row  = lane & 15;        // M for A-frags / N within tile for B,C,D
  const int koff = (lane >> 4) * 2;  // K sub-offset per lane half
  const int mbase = (lane >> 4) * 8; // D-tile M base per lane half

  // ---- preload A fragments for matmul1 (16 K-steps of 4) ----
  v2f afrag[16];
#pragma unroll
  for (int kk = 0; kk < 16; ++kk)
    afrag[kk] = *(const v2f*)(slice + row * DIM + 4 * kk + koff);

  v8f acc[4];

  // ---- matmul1: (x+agg) @ W1, +b1, relu ----
#pragma unroll
  for (int t = 0; t < 4; ++t) {
    v8f c = {0.f, 0.f, 0.f, 0.f, 0.f, 0.f, 0.f, 0.f};
#pragma unroll
    for (int kk = 0; kk < 16; ++kk) {
      int k = 4 * kk + koff;
      v2f b;
      b.x = lds_w[k * DIM + t * 16 + row];
      b.y = lds_w[(k + 1) * DIM + t * 16 + row];
      c = __builtin_amdgcn_wmma_f32_16x16x4_f32(false, afrag[kk], false, b,
                                                (short)0, c, false, false);
    }
    float bb = b1[t * 16 + row];
#pragma unroll
    for (int v = 0; v < 8; ++v) {
      float h = c[v] + bb;
      c[v] = h > 0.f ? h : 0.f;
    }
    acc[t] = c;
  }

  // transpose h1 through wave-private LDS slice (D layout -> A layout)
#pragma unroll
  for (int t = 0; t < 4; ++t)
#pragma unroll
    for (int v = 0; v < 8; ++v)
      slice[(mbase + v) * DIM + t * 16 + row] = acc[t][v];

#pragma unroll
  for (int kk = 0; kk < 16; ++kk)
    afrag[kk] = *(const v2f*)(slice + row * DIM + 4 * kk + koff);

  // ---- matmul2: h1 @ W2, +b2, BN, relu ----
  const float* W2l = lds_w + DIM * DIM;
#pragma unroll
  for (int t = 0; t < 4; ++t) {
    v8f c = {0.f, 0.f, 0.f, 0.f, 0.f, 0.f, 0.f, 0.f};
#pragma unroll
    for (int kk = 0; kk < 16; ++kk) {
      int k = 4 * kk + koff;
      v2f b;
      b.x = W2l[k * DIM + t * 16 + row];
      b.y = W2l[(k + 1) * DIM + t * 16 + row];
      c = __builtin_amdgcn_wmma_f32_16x16x4_f32(false, afrag[kk], false, b,
                                                (short)0, c, false, false);
    }
    int col = t * 16 + row;
    float bb = b2[col];
    float sc = gamma[col] * rsqrtf(var[col] + BN_EPS);
    float sh = beta[col] - mean[col] * sc;
#pragma unroll
    for (int v = 0; v < 8; ++v) {
      float h = (c[v] + bb) * sc + sh;
      acc[t][v] = h > 0.f ? h : 0.f;
    }
  }

  // stage final tile for coalesced output
#pragma unroll
  for (int t = 0; t < 4; ++t)
#pragma unroll
    for (int v = 0; v < 8; ++v)
      slice[(mbase + v) * DIM + t * 16 + row] = acc[t][v];
  __syncthreads();

  for (int i = tid; i < 64 * DIM / 4; i += 128) {
    int node = block_node0 + (i >> 4);
    if (node < N_NODES)
      ((float4*)(xout + (size_t)node * DIM))[i & 15] = ((const float4*)lds_tile)[i];
  }
}

// g[batch[n]] += x[n]
__global__ void pool_kernel(const float* __restrict__ x,
                            const int* __restrict__ batch,
                            float* __restrict__ g) {
  int gid = blockIdx.x * blockDim.x + threadIdx.x;
  if (gid >= N_NODES * 16) return;
  int n = gid >> 4;
  int c = (gid & 15) * 4;
  int b = batch[n];
  const float4 v = *(const float4*)(x + (size_t)n * DIM + c);
  float* dst = g + (size_t)b * DIM + c;
  atomicAdd(dst + 0, v.x);
  atomicAdd(dst + 1, v.y);
  atomicAdd(dst + 2, v.z);
  atomicAdd(dst + 3, v.w);
}

// out[g] = relu([g1|g2|kge[rel]] @ Wf1 + bf1) @ Wf2 + bf2   (tiny head)
__global__ void final_kernel(const float* __restrict__ g1,
                             const float* __restrict__ g2,
                             const int* __restrict__ rel,
                             const float* __restrict__ kge,
                             const float* __restrict__ Wf1,  // (256,64)
                             const float* __restrict__ bf1,
                             const float* __restrict__ Wf2,  // (64,1)
                             const float* __restrict__ bf2,
                             float* __restrict__ out) {
  __shared__ float z[2 * DIM + KGE_DIM];
  __shared__ float h[DIM];
  int g = blockIdx.x;
  int tid = threadIdx.x;
  if (tid < DIM)            z[tid] = g1[g * DIM + tid];
  else if (tid < 2 * DIM)   z[tid] = g2[g * DIM + (tid - DIM)];
  else                      z[tid] = kge[(size_t)rel[g] * KGE_DIM + (tid - 2 * DIM)];
  __syncthreads();
  if (tid < DIM) {
    float s = bf1[tid];
#pragma unroll 8
    for (int k = 0; k < 2 * DIM + KGE_DIM; ++k) s += z[k] * Wf1[k * DIM + tid];
    h[tid] = s > 0.f ? s : 0.f;
  }
  __syncthreads();
  if (tid == 0) {
    float s = bf2[0];
#pragma unroll 8
    for (int j = 0; j < DIM; ++j) s += h[j] * Wf2[j];
    out[g] = s;
  }
}

extern "C" void kernel_launch(void* const* d_in, const int* in_sizes, int n_in,
                              void* d_out, int out_size, void* d_ws, size_t ws_size,
                              hipStream_t stream) {
  const float* x1     = (const float*)d_in[0];
  const int*   ei1    = (const int*)d_in[1];
  const int*   batch1 = (const int*)d_in[2];
  const float* x2     = (const float*)d_in[3];
  const int*   ei2    = (const int*)d_in[4];
  const int*   batch2 = (const int*)d_in[5];
  const int*   rel    = (const int*)d_in[6];
  const float* W1     = (const float*)d_in[7];
  const float* b1     = (const float*)d_in[8];
  const float* W2     = (const float*)d_in[9];
  const float* b2     = (const float*)d_in[10];
  const float* gamma  = (const float*)d_in[11];
  const float* beta   = (const float*)d_in[12];
  const float* mean   = (const float*)d_in[13];
  const float* var    = (const float*)d_in[14];
  const float* kge    = (const float*)d_in[15];
  const float* Wf1    = (const float*)d_in[16];
  const float* bf1    = (const float*)d_in[17];
  const float* Wf2    = (const float*)d_in[18];
  const float* bf2    = (const float*)d_in[19];
  float* out = (float*)d_out;

  float* ws   = (float*)d_ws;
  float* bufA = ws;
  float* bufB = bufA + (size_t)N_NODES * DIM;
  float* agg  = bufB + (size_t)N_NODES * DIM;
  float* g1   = agg  + (size_t)N_NODES * DIM;
  float* g2   = g1 + (size_t)N_GRAPHS * DIM;

  const int edge_blocks = (N_EDGES * 16 + 255) / 256;          // 50000
  const int nvec_blocks = (N_NODES * 16 + 255) / 256;          // 3125
  const int zero_blocks = (N_NODES * DIM + 255) / 256;
  const int mlp_blocks  = (N_NODES + 63) / 64;                 // 782

  for (int p = 0; p < 2; ++p) {
    const float* xin   = (p == 0) ? x1 : x2;
    const int*   ei    = (p == 0) ? ei1 : ei2;
    const int*   batch = (p == 0) ? batch1 : batch2;
    float*       gout  = (p == 0) ? g1 : g2;
    const float* xc = xin;
    for (int l = 0; l < 3; ++l) {
      float* xn = (l & 1) ? bufB : bufA;
      zero_kernel<<<zero_blocks, 256, 0, stream>>>(agg, N_NODES * DIM);
      edge_agg_kernel<<<edge_blocks, 256, 0, stream>>>(xc, ei, agg);
      node_mlp_kernel<<<mlp_blocks, 128, 0, stream>>>(
          xc, agg, W1 + l * DIM * DIM, b1 + l * DIM, W2 + l * DIM * DIM,
          b2 + l * DIM, gamma + l * DIM, beta + l * DIM, mean + l * DIM,
          var + l * DIM, xn);
      xc = xn;
    }
    zero_kernel<<<(N_GRAPHS * DIM + 255) / 256, 256, 0, stream>>>(gout, N_GRAPHS * DIM);
    pool_kernel<<<nvec_blocks, 256, 0, stream>>>(xc, batch, gout);
  }
  final_kernel<<<N_GRAPHS, 256, 0, stream>>>(g1, g2, rel, kge, Wf1, bf1, Wf2, bf2, out);
}